// SphericalSpectralTimeConv1D_38405597561166
// MI455X (gfx1250) — compile-verified
//
#include <hip/hip_runtime.h>
#include <hip/hip_bf16.h>

// ---------------------------------------------------------------------------
// SphericalSpectralTimeConv1D on MI455X (gfx1250)
// L=64, NLAT=64, NLON=127, C_IN=C_OUT=64, T_EMB=128
// Dominant cost: streaming 532 MB of dense weights (GEMV) ~= 23 us @ 23.3TB/s.
// All spectral GEMMs use V_WMMA_F32_16X16X4_F32 (full f32 precision).
// ---------------------------------------------------------------------------

#define LL      64
#define NLON    127
#define CC      64
#define TEMB    128
#define PLANE   (LL * NLON * CC)      // 520192 floats per (l,m,c) plane
#define ROWS    (NLON * CC)           // 8128: stride between l (or t) rows
#define N4      (PLANE / 4)           // 130048 float4 per dense plane

typedef float v2f __attribute__((ext_vector_type(2)));
typedef float v4f __attribute__((ext_vector_type(4)));
typedef float v8f __attribute__((ext_vector_type(8)));

static __device__ __forceinline__ v8f wmma4(v2f a, v2f b, v8f c) {
  // D = A(16x4 f32) * B(4x16 f32) + C(16x16 f32)
  return __builtin_amdgcn_wmma_f32_16x16x4_f32(
      false, a, false, b, (short)0, c, false, false);
}

// A fragment: element (r0 + lane%16, k0 + 2*(lane/16) + {0,1})
static __device__ __forceinline__ v2f loadA(const float* __restrict__ base,
                                            int r0, int k0, int rs, int cs,
                                            int lane, int rmax) {
  int r = r0 + (lane & 15);
  r = (r > rmax) ? rmax : r;
  int k = k0 + 2 * (lane >> 4);
  v2f a;
  a.x = base[(size_t)r * rs + (size_t)k * cs];
  a.y = base[(size_t)r * rs + (size_t)(k + 1) * cs];
  return a;
}

// B fragment: element (k0 + 2*(lane/16) + {0,1}, n0 + lane%16)
static __device__ __forceinline__ v2f loadB(const float* __restrict__ base,
                                            int k0, int n0, int rs, int cs,
                                            int lane, int kmax) {
  int n = n0 + (lane & 15);
  int k = k0 + 2 * (lane >> 4);
  int ka = (k > kmax) ? kmax : k;
  int kb = (k + 1 > kmax) ? kmax : (k + 1);
  v2f b;
  b.x = base[(size_t)ka * rs + (size_t)n * cs];
  b.y = base[(size_t)kb * rs + (size_t)n * cs];
  return b;
}

// C/D store: lane, vgpr v -> row r0 + v + 8*(lane/16), col c0 + lane%16
static __device__ __forceinline__ void storeC(float* __restrict__ base,
                                              int r0, int c0, int rs, int cs,
                                              int lane, v8f acc, int rmax) {
  int c = c0 + (lane & 15);
#pragma unroll
  for (int v = 0; v < 8; ++v) {
    int r = r0 + v + 8 * (lane >> 4);
    if (r <= rmax) base[(size_t)r * rs + (size_t)c * cs] = acc[v];
  }
}

// ---------------------------------------------------------------------------
// Kernel 1: Gauss-Legendre nodes/weights + normalized associated Legendre
// tables Pfull[l, j, t] (j = 63 +/- m) and PW = Pfull * wg[t].  One thread
// per quadrature node t; all in double precision.
// ---------------------------------------------------------------------------
__global__ void k_legendre(float* __restrict__ Pfull, float* __restrict__ PW,
                           float* __restrict__ wgout) {
  const double PI_D = 3.14159265358979323846;
  int t = threadIdx.x;
  if (t >= LL) return;
  const int n = LL;
  // ascending nodes: i = n - t gives x_t increasing with t
  double x = cos(PI_D * ((double)(n - t) - 0.25) / ((double)n + 0.5));
  double pp = 1.0;
  for (int it = 0; it < 64; ++it) {
    double p0 = 1.0, p1 = x;
    for (int k = 2; k <= n; ++k) {
      double p2 = ((2.0 * k - 1.0) * x * p1 - (k - 1.0) * p0) / (double)k;
      p0 = p1; p1 = p2;
    }
    pp = (double)n * (x * p1 - p0) / (x * x - 1.0);
    x -= p1 / pp;
  }
  { // final derivative at converged x
    double p0 = 1.0, p1 = x;
    for (int k = 2; k <= n; ++k) {
      double p2 = ((2.0 * k - 1.0) * x * p1 - (k - 1.0) * p0) / (double)k;
      p0 = p1; p1 = p2;
    }
    pp = (double)n * (x * p1 - p0) / (x * x - 1.0);
  }
  double w = 2.0 / ((1.0 - x * x) * pp * pp);
  wgout[t] = (float)w;

  double sx = sqrt(fmax(1.0 - x * x, 0.0));
  double pdiag = sqrt(1.0 / (4.0 * PI_D));        // Pbar_00
  for (int m = 0; m < LL; ++m) {
    if (m > 0)
      pdiag = -sqrt((2.0 * m + 1.0) / (2.0 * m)) * sx * pdiag;
    int jp = (LL - 1) + m, jm = (LL - 1) - m;
    double sgn = (m & 1) ? -1.0 : 1.0;
    double plm2 = 0.0, plm1 = 0.0;
    for (int l = 0; l < LL; ++l) {
      double val;
      if (l < m) {
        val = 0.0;
      } else if (l == m) {
        val = pdiag;
      } else if (l == m + 1) {
        val = sqrt(2.0 * m + 3.0) * x * pdiag;
      } else {
        double a = sqrt((4.0 * (double)l * l - 1.0) /
                        ((double)l * l - (double)m * m));
        double b = sqrt((((double)l - 1.0) * ((double)l - 1.0) - (double)m * m) /
                        (4.0 * ((double)l - 1.0) * ((double)l - 1.0) - 1.0));
        val = a * (x * plm1 - b * plm2);
      }
      if (l >= m) { plm2 = plm1; plm1 = val; }
      float fv = (float)val;
      size_t ip = ((size_t)l * NLON + jp) * CC + t;
      Pfull[ip] = fv;
      PW[ip] = fv * (float)w;
      if (m > 0) {
        size_t im = ((size_t)l * NLON + jm) * CC + t;
        float fvm = (float)(sgn * val);
        Pfull[im] = fvm;
        PW[im] = fvm * (float)w;
      }
    }
  }
}

// ---------------------------------------------------------------------------
// Kernel 2: zero-padded 128x128 DFT matrices.
//   forward  F[j][p] = (2*pi/127) * exp(-2*pi*i*(j-63)*p/127)
//   inverse  G[p][j] =              exp(+2*pi*i*(j-63)*p/127)
// ---------------------------------------------------------------------------
__global__ void k_fourier(float* __restrict__ Ffr, float* __restrict__ Ffi,
                          float* __restrict__ Gr, float* __restrict__ Gi) {
  const double PI_D = 3.14159265358979323846;
  int j = blockIdx.x;    // 0..127
  int p = threadIdx.x;   // 0..127
  size_t idx = (size_t)j * 128 + p;
  if (j >= NLON || p >= NLON) {
    Ffr[idx] = 0.0f; Ffi[idx] = 0.0f; Gr[idx] = 0.0f; Gi[idx] = 0.0f;
    return;
  }
  int mj = j - (LL - 1);
  int r = (mj * p) % NLON;
  if (r < 0) r += NLON;
  double ang = 2.0 * PI_D * (double)r / (double)NLON;
  double s, c;
  sincos(ang, &s, &c);
  double scale = 2.0 * PI_D / (double)NLON;
  Ffr[idx] = (float)(scale * c);
  Ffi[idx] = (float)(-scale * s);
  size_t gidx = (size_t)p * 128 + j;
  Gr[gidx] = (float)c;
  Gi[gidx] = (float)s;
}

// ---------------------------------------------------------------------------
// Kernel 3: t_c = t_emb @ dense_w  (the bandwidth-dominant stage: 532 MB).
// Each thread owns one float4 column slice, streams 128 rows of both planes
// with non-temporal b128 loads (532 MB >> 192 MB L2; don't pollute it).
// ---------------------------------------------------------------------------
__global__ void k_temb_gemv(const float* __restrict__ temb,
                            const float* __restrict__ dwr,
                            const float* __restrict__ dwi,
                            float* __restrict__ tcr, float* __restrict__ tci) {
  __shared__ float te[TEMB];
  int tid = threadIdx.x;
  if (tid < TEMB) te[tid] = temb[tid];
  __syncthreads();
  int n4 = blockIdx.x * blockDim.x + tid;
  if (n4 >= N4) return;
  const v4f* __restrict__ pr = (const v4f*)dwr;
  const v4f* __restrict__ pi = (const v4f*)dwi;
  v4f ar = {}, ai = {};
#pragma unroll 4
  for (int k = 0; k < TEMB; ++k) {
    float tk = te[k];
    v4f vr = __builtin_nontemporal_load(&pr[(size_t)k * N4 + n4]);
    v4f vi = __builtin_nontemporal_load(&pi[(size_t)k * N4 + n4]);
    ar += tk * vr;
    ai += tk * vi;
  }
  ((v4f*)tcr)[n4] = ar;
  ((v4f*)tci)[n4] = ai;
}

// ---------------------------------------------------------------------------
// Kernel 4: forward DFT.  fm[t,j,c] = sum_p F[j,p] * x[t,p,c]   (A complex,
// B real).  Grid (t=64, jtile=8); 4 waves/block, one c-tile each.
// ---------------------------------------------------------------------------
__global__ void k_fwd_dft(const float* __restrict__ x,
                          const float* __restrict__ Ffr,
                          const float* __restrict__ Ffi,
                          float* __restrict__ fmr, float* __restrict__ fmi) {
  int t = blockIdx.x;
  int j0 = blockIdx.y * 16;
  int lane = threadIdx.x & 31;
  int c0 = (threadIdx.x >> 5) * 16;
  const float* xb = x + (size_t)t * ROWS;      // (p,c) rs=64
  v8f ar = {}, ai = {};
  for (int k0 = 0; k0 < 128; k0 += 4) {
    v2f far = loadA(Ffr, j0, k0, 128, 1, lane, 127);
    v2f fai = loadA(Ffi, j0, k0, 128, 1, lane, 127);
    v2f bx  = loadB(xb, k0, c0, CC, 1, lane, NLON - 1);
    ar = wmma4(far, bx, ar);
    ai = wmma4(fai, bx, ai);
  }
  storeC(fmr + (size_t)t * ROWS, j0, c0, CC, 1, lane, ar, NLON - 1);
  storeC(fmi + (size_t)t * ROWS, j0, c0, CC, 1, lane, ai, NLON - 1);
}

// ---------------------------------------------------------------------------
// Kernel 5: Legendre transform.  x_sht[l,m,c] = sum_t PW[l,m,t]*fm[t,m,c]
// (A real, B complex).  Grid (m=127, ltile=4); 4 waves = c-tiles.
// ---------------------------------------------------------------------------
__global__ void k_sht(const float* __restrict__ PW,
                      const float* __restrict__ fmr,
                      const float* __restrict__ fmi,
                      float* __restrict__ shtr, float* __restrict__ shti) {
  int m = blockIdx.x;
  int l0 = blockIdx.y * 16;
  int lane = threadIdx.x & 31;
  int c0 = (threadIdx.x >> 5) * 16;
  const float* A  = PW  + (size_t)m * CC;   // (l,t): rs=ROWS, cs=1
  const float* Br = fmr + (size_t)m * CC;   // (t,c): rs=ROWS, cs=1
  const float* Bi = fmi + (size_t)m * CC;
  v8f ar = {}, ai = {};
  for (int k0 = 0; k0 < CC; k0 += 4) {
    v2f a  = loadA(A, l0, k0, ROWS, 1, lane, LL - 1);
    v2f br = loadB(Br, k0, c0, ROWS, 1, lane, LL - 1);
    v2f bi = loadB(Bi, k0, c0, ROWS, 1, lane, LL - 1);
    ar = wmma4(a, br, ar);
    ai = wmma4(a, bi, ai);
  }
  storeC(shtr + (size_t)m * CC, l0, c0, ROWS, 1, lane, ar, LL - 1);
  storeC(shti + (size_t)m * CC, l0, c0, ROWS, 1, lane, ai, LL - 1);
}

// ---------------------------------------------------------------------------
// Kernel 6: spectral conv + time embedding.
// res[l,m,o] = t_c[l,m,o] + sum_i x_sht[l,m,i] * W[l,i,o]  (complex*complex).
// t_c is loaded directly as the WMMA C operand.  Grid (l=64, mtile=8).
// ---------------------------------------------------------------------------
__global__ void k_conv(const float* __restrict__ shtr,
                       const float* __restrict__ shti,
                       const float* __restrict__ wr,
                       const float* __restrict__ wi,
                       const float* __restrict__ tcr,
                       const float* __restrict__ tci,
                       float* __restrict__ resr, float* __restrict__ resi) {
  int l = blockIdx.x;
  int m0 = blockIdx.y * 16;
  int lane = threadIdx.x & 31;
  int o0 = (threadIdx.x >> 5) * 16;
  const float* Ar = shtr + (size_t)l * ROWS;     // (m,i): rs=64
  const float* Ai = shti + (size_t)l * ROWS;
  const float* Br = wr + (size_t)l * CC * CC;    // (i,o): rs=64
  const float* Bi = wi + (size_t)l * CC * CC;
  v8f ar, ai;
  { // initialize accumulators with t_c tile
    int o = o0 + (lane & 15);
#pragma unroll
    for (int v = 0; v < 8; ++v) {
      int m = m0 + v + 8 * (lane >> 4);
      int mc = (m > NLON - 1) ? (NLON - 1) : m;
      size_t idx = (size_t)l * ROWS + (size_t)mc * CC + o;
      ar[v] = tcr[idx];
      ai[v] = tci[idx];
    }
  }
  for (int k0 = 0; k0 < CC; k0 += 4) {
    v2f xr = loadA(Ar, m0, k0, CC, 1, lane, NLON - 1);
    v2f xi = loadA(Ai, m0, k0, CC, 1, lane, NLON - 1);
    v2f br = loadB(Br, k0, o0, CC, 1, lane, CC - 1);
    v2f bi = loadB(Bi, k0, o0, CC, 1, lane, CC - 1);
    v2f nxi = -xi;
    ar = wmma4(xr, br, ar);
    ar = wmma4(nxi, bi, ar);
    ai = wmma4(xr, bi, ai);
    ai = wmma4(xi, br, ai);
  }
  storeC(resr + (size_t)l * ROWS, m0, o0, CC, 1, lane, ar, NLON - 1);
  storeC(resi + (size_t)l * ROWS, m0, o0, CC, 1, lane, ai, NLON - 1);
}

// ---------------------------------------------------------------------------
// Kernel 7: inverse Legendre.  fm_out[t,m,c] = sum_l Pfull[l,m,t]*res[l,m,c]
// A = Pfull^T per m: (t,l) with rs=1, cs=ROWS.  Grid (m=127, ttile=4).
// ---------------------------------------------------------------------------
__global__ void k_isht(const float* __restrict__ Pfull,
                       const float* __restrict__ resr,
                       const float* __restrict__ resi,
                       float* __restrict__ fmor, float* __restrict__ fmoi) {
  int m = blockIdx.x;
  int t0 = blockIdx.y * 16;
  int lane = threadIdx.x & 31;
  int c0 = (threadIdx.x >> 5) * 16;
  const float* A  = Pfull + (size_t)m * CC;  // (t,l): rs=1, cs=ROWS
  const float* Br = resr + (size_t)m * CC;   // (l,c): rs=ROWS, cs=1
  const float* Bi = resi + (size_t)m * CC;
  v8f ar = {}, ai = {};
  for (int k0 = 0; k0 < CC; k0 += 4) {
    v2f a  = loadA(A, t0, k0, 1, ROWS, lane, LL - 1);
    v2f br = loadB(Br, k0, c0, ROWS, 1, lane, LL - 1);
    v2f bi = loadB(Bi, k0, c0, ROWS, 1, lane, LL - 1);
    ar = wmma4(a, br, ar);
    ai = wmma4(a, bi, ai);
  }
  storeC(fmor + (size_t)m * CC, t0, c0, ROWS, 1, lane, ar, LL - 1);
  storeC(fmoi + (size_t)m * CC, t0, c0, ROWS, 1, lane, ai, LL - 1);
}

// ---------------------------------------------------------------------------
// Kernel 8: inverse DFT + transpose to (c, t, p) interleaved complex output.
// out[c,t,p] = sum_j G[p,j] * fm_out[t,j,c]   (complex*complex).
// ---------------------------------------------------------------------------
__global__ void k_inv_dft(const float* __restrict__ fmor,
                          const float* __restrict__ fmoi,
                          const float* __restrict__ Gr,
                          const float* __restrict__ Gi,
                          float* __restrict__ out) {
  int t = blockIdx.x;
  int p0 = blockIdx.y * 16;
  int lane = threadIdx.x & 31;
  int c0 = (threadIdx.x >> 5) * 16;
  const float* Br = fmor + (size_t)t * ROWS;   // (j,c): rs=64
  const float* Bi = fmoi + (size_t)t * ROWS;
  v8f ar = {}, ai = {};
  for (int k0 = 0; k0 < 128; k0 += 4) {
    v2f gr = loadA(Gr, p0, k0, 128, 1, lane, 127);
    v2f gi = loadA(Gi, p0, k0, 128, 1, lane, 127);
    v2f br = loadB(Br, k0, c0, CC, 1, lane, NLON - 1);
    v2f bi = loadB(Bi, k0, c0, CC, 1, lane, NLON - 1);
    v2f ngi = -gi;
    ar = wmma4(gr, br, ar);
    ar = wmma4(ngi, bi, ar);
    ai = wmma4(gr, bi, ai);
    ai = wmma4(gi, br, ai);
  }
  int c = c0 + (lane & 15);
#pragma unroll
  for (int v = 0; v < 8; ++v) {
    int p = p0 + v + 8 * (lane >> 4);
    if (p < NLON) {
      size_t idx = (((size_t)c * LL + t) * NLON + p) * 2;
      out[idx]     = ar[v];
      out[idx + 1] = ai[v];
    }
  }
}

// ---------------------------------------------------------------------------
extern "C" void kernel_launch(void* const* d_in, const int* in_sizes, int n_in,
                              void* d_out, int out_size, void* d_ws,
                              size_t ws_size, hipStream_t stream) {
  const float* x    = (const float*)d_in[0];  // (64,127,64)
  const float* temb = (const float*)d_in[1];  // (128,)
  const float* wr   = (const float*)d_in[2];  // (64,64,64)
  const float* wi   = (const float*)d_in[3];  // (64,64,64)
  const float* dwr  = (const float*)d_in[4];  // (128, 520192)
  const float* dwi  = (const float*)d_in[5];  // (128, 520192)
  float* out = (float*)d_out;                 // (64,64,127) complex interleaved
  float* ws = (float*)d_ws;

  float* Pfull = ws;                 // PLANE
  float* PW    = Pfull + PLANE;
  float* tcr   = PW + PLANE;
  float* tci   = tcr + PLANE;
  float* fmr   = tci + PLANE;
  float* fmi   = fmr + PLANE;
  float* shtr  = fmi + PLANE;
  float* shti  = shtr + PLANE;
  float* resr  = shti + PLANE;
  float* resi  = resr + PLANE;
  float* fmor  = resi + PLANE;
  float* fmoi  = fmor + PLANE;
  float* Ffr   = fmoi + PLANE;       // 128*128 each, zero padded
  float* Ffi   = Ffr + 128 * 128;
  float* Gr    = Ffi + 128 * 128;
  float* Gi    = Gr + 128 * 128;
  float* wg    = Gi + 128 * 128;     // 64

  k_legendre<<<dim3(1), dim3(64), 0, stream>>>(Pfull, PW, wg);
  k_fourier<<<dim3(128), dim3(128), 0, stream>>>(Ffr, Ffi, Gr, Gi);
  k_temb_gemv<<<dim3(N4 / 256), dim3(256), 0, stream>>>(temb, dwr, dwi, tcr, tci);
  k_fwd_dft<<<dim3(64, 8), dim3(128), 0, stream>>>(x, Ffr, Ffi, fmr, fmi);
  k_sht<<<dim3(127, 4), dim3(128), 0, stream>>>(PW, fmr, fmi, shtr, shti);
  k_conv<<<dim3(64, 8), dim3(128), 0, stream>>>(shtr, shti, wr, wi, tcr, tci,
                                                resr, resi);
  k_isht<<<dim3(127, 4), dim3(128), 0, stream>>>(Pfull, resr, resi, fmor, fmoi);
  k_inv_dft<<<dim3(64, 8), dim3(128), 0, stream>>>(fmor, fmoi, Gr, Gi, out);
}